// Model_81905026335186
// MI455X (gfx1250) — compile-verified
//
#include <hip/hip_runtime.h>

// Problem constants (match reference)
#define Bn 16
#define Tn 2048
#define Sn 512
#define Cn 192
#define NEGV (-1e9f)
#define LOG2PI 1.8378770664093453f

typedef __attribute__((ext_vector_type(2))) float v2f;
typedef __attribute__((ext_vector_type(8))) float v8f;

// ---------------------------------------------------------------------------
// Prep 1: per (b,s,c):
//   P12t[b,c,s] = { m*exp(-2*logs_p), -0.5*exp(-2*logs_p) }   (interleaved float2
//   so a WMMA B-fragment is a single global_load_b64)
//   nc14[b,s]   = sum_c( -0.5*log(2pi) - logs_p - 0.5*m^2*exp(-2*logs_p) )
// ---------------------------------------------------------------------------
__global__ __launch_bounds__(256) void prep_stats(const float* __restrict__ m_p,
                                                  const float* __restrict__ logs_p,
                                                  float* __restrict__ P12t,
                                                  float* __restrict__ nc14) {
  const int bs = blockIdx.x;
  const int b = bs / Sn, s = bs % Sn;
  const int c = threadIdx.x;
  float term = 0.0f;
  if (c < Cn) {
    const size_t i = ((size_t)b * Sn + s) * Cn + c;
    const float lp = logs_p[i];
    const float m  = m_p[i];
    const float ssr = __expf(-2.0f * lp);
    float2 pk;
    pk.x = m * ssr;
    pk.y = -0.5f * ssr;
    *(float2*)(P12t + (((size_t)b * Cn + c) * Sn + s) * 2) = pk;
    term = -0.5f * LOG2PI - lp - 0.5f * m * m * ssr;
  }
  __shared__ float red[256];
  red[threadIdx.x] = term;
  __syncthreads();
  for (int off = 128; off > 0; off >>= 1) {
    if ((int)threadIdx.x < off) red[threadIdx.x] += red[threadIdx.x + off];
    __syncthreads();
  }
  if (threadIdx.x == 0) nc14[(size_t)b * Sn + s] = red[0];
}

// ---------------------------------------------------------------------------
// Prep 2: transpose z_p [B,T,C] -> zt [B,C,T] (coalesced A-fragment loads)
// ---------------------------------------------------------------------------
__global__ __launch_bounds__(256) void prep_zt(const float* __restrict__ z_p,
                                               float* __restrict__ zt) {
  const size_t idx = (size_t)blockIdx.x * 256 + threadIdx.x;  // B*T*C total
  const int c = (int)(idx % Cn);
  const size_t bt = idx / Cn;
  const int t = (int)(bt % Tn);
  const int b = (int)(bt / Tn);
  zt[((size_t)b * Cn + c) * Tn + t] = z_p[idx];
}

// ---------------------------------------------------------------------------
// neg_cent [B,T,S] via fp32 WMMA (V_WMMA_F32_16X16X4_F32), K = 2C = 384.
// Even k rows: z * (m*ssr)  (nc3);  odd k rows: z^2 * (-0.5*ssr)  (nc2).
// Each wave: one 16(t) x 64(s) strip (4 accumulators, A reused 4x).
// Block = 4 waves -> 64x64 macro tile.  Result written to d_out attn region.
// ---------------------------------------------------------------------------
__global__ __launch_bounds__(128) void negcent_wmma(const float* __restrict__ zt,   // [B,C,T]
                                                    const float* __restrict__ P12t, // [B,C,S] float2
                                                    const float* __restrict__ nc14, // [B,S]
                                                    float* __restrict__ out) {      // [B,T,S]
  const int b    = blockIdx.z;
  const int lane = threadIdx.x & 31;
  const int wave = threadIdx.x >> 5;
  const int t0   = blockIdx.y * 64 + wave * 16;
  const int s0   = blockIdx.x * 64;
  const int half = lane >> 4;   // 0: K pair {4q,4q+1}=chan 2q ; 1: {4q+2,4q+3}=chan 2q+1
  const int l15  = lane & 15;

  const float* zb   = zt   + (size_t)b * Cn * Tn;
  const float* p12b = P12t + (size_t)b * Cn * Sn * 2;

  v8f acc[4];
  acc[0] = v8f{}; acc[1] = v8f{}; acc[2] = v8f{}; acc[3] = v8f{};

  for (int q = 0; q < (2 * Cn) / 4; ++q) {   // 96 k-steps of 4
    const int c = 2 * q + half;
    const float z = zb[(size_t)c * Tn + t0 + l15];
    v2f a; a.x = z; a.y = z * z;
    const v2f* p12c = (const v2f*)(p12b + ((size_t)c * Sn + s0 + l15) * 2);
#pragma unroll
    for (int j = 0; j < 4; ++j) {
      const v2f bf = p12c[16 * j];   // one global_load_b64 = {P1, P2}
      acc[j] = __builtin_amdgcn_wmma_f32_16x16x4_f32(false, a, false, bf,
                                                     (short)0, acc[j], false, false);
    }
  }

  // C/D layout: VGPR r, lanes 0-15 -> M=r, lanes 16-31 -> M=8+r ; N = lane&15
  const int trow = t0 + 8 * half;
  float* ob = out + (size_t)b * Tn * Sn;
#pragma unroll
  for (int j = 0; j < 4; ++j) {
    const int s = s0 + 16 * j + l15;
    const float nc = nc14[(size_t)b * Sn + s];
#pragma unroll
    for (int r = 0; r < 8; ++r) {
      ob[(size_t)(trow + r) * Sn + s] = acc[j][r] + nc;
    }
  }
}

// ---------------------------------------------------------------------------
// MAS forward: Viterbi scan over T. One 512-thread block per batch.
// stay bits packed with wave32 ballot into dir[B][T][S/32].
// ---------------------------------------------------------------------------
__global__ __launch_bounds__(512) void mas_forward(const float* __restrict__ negc,
                                                   const int* __restrict__ textlen,
                                                   const int* __restrict__ mellen,
                                                   unsigned* __restrict__ dir) {
  const int b = blockIdx.x;
  const int s = threadIdx.x;
  __shared__ float buf[Sn];
  const int tl = textlen[b];
  const int ml = mellen[b];
  const float* nb = negc + (size_t)b * Tn * Sn;
  unsigned* db = dir + (size_t)b * Tn * (Sn / 32);

  float p = 0.0f;
  for (int t = 0; t < Tn; ++t) {
    buf[s] = p;
    __syncthreads();
    const float prev = (s == 0) ? NEGV : buf[s - 1];
    const bool stay = (p >= prev);
    const unsigned mask = __builtin_amdgcn_ballot_w32(stay);
    if ((s & 31) == 0) db[(size_t)t * (Sn / 32) + (s >> 5)] = mask;
    const float ll = (s < tl && t < ml) ? nb[(size_t)t * Sn + s] : NEGV;
    const float best = stay ? p : prev;
    p = (s <= t) ? (best + ll) : NEGV;
    __syncthreads();
  }
}

// ---------------------------------------------------------------------------
// MAS backward: serial per batch; records path index seq[b,t] (-1 if inactive).
// Also computes denom = sum(mellen).
// ---------------------------------------------------------------------------
__global__ void mas_backward(const int* __restrict__ textlen,
                             const int* __restrict__ mellen,
                             const unsigned* __restrict__ dir,
                             int* __restrict__ seq,
                             float* __restrict__ denom) {
  const int b = threadIdx.x;
  if (b < Bn) {
    int idx = textlen[b] - 1;
    const int ml = mellen[b];
    const unsigned* db = dir + (size_t)b * Tn * (Sn / 32);
    for (int t = Tn - 1; t >= 0; --t) {
      const bool active = (t < ml);
      seq[b * Tn + t] = active ? idx : -1;
      const int ridx = (idx < 0) ? 0 : ((idx > Sn - 1) ? Sn - 1 : idx);
      const unsigned w = db[(size_t)t * (Sn / 32) + (ridx >> 5)];
      const int stay = (w >> (ridx & 31)) & 1;
      if (active) idx -= (1 - stay);
    }
  }
  if (threadIdx.x == 0) {
    float d = 0.0f;
    for (int i = 0; i < Bn; ++i) d += (float)mellen[i];
    *denom = d;
  }
}

// ---------------------------------------------------------------------------
// attn one-hot write + KL gather/reduce. One 256-thread block per (b,t).
// Overwrites the neg_cent scratch with the final attn output.
// ---------------------------------------------------------------------------
__global__ __launch_bounds__(256) void kl_attn(const float* __restrict__ m_p,
                                               const float* __restrict__ logs_p,
                                               const float* __restrict__ z_p,
                                               const float* __restrict__ logs_q,
                                               const int* __restrict__ seq,
                                               float* __restrict__ attn,
                                               float* __restrict__ acc) {
  const int t = blockIdx.x;
  const int b = blockIdx.y;
  const int sidx = seq[b * Tn + t];

  // attn row: one-hot (all zero when inactive)
  float* arow = attn + ((size_t)b * Tn + t) * Sn;
  const int s2 = threadIdx.x * 2;
  float2 w;
  w.x = (s2 == sidx) ? 1.0f : 0.0f;
  w.y = (s2 + 1 == sidx) ? 1.0f : 0.0f;
  *(float2*)(arow + s2) = w;

  // KL contribution
  float term = 0.0f;
  const int c = threadIdx.x;
  if (sidx >= 0 && c < Cn) {
    const size_t pi = ((size_t)b * Sn + sidx) * Cn + c;
    const size_t qi = ((size_t)b * Tn + t) * Cn + c;
    const float me = m_p[pi];
    const float le = logs_p[pi];
    const float z  = z_p[qi];
    const float lq = logs_q[qi];
    const float d  = z - me;
    term = le - lq - 0.5f + 0.5f * d * d * __expf(-2.0f * le);
  }
  __shared__ float red[256];
  red[threadIdx.x] = term;
  __syncthreads();
  for (int off = 128; off > 0; off >>= 1) {
    if ((int)threadIdx.x < off) red[threadIdx.x] += red[threadIdx.x + off];
    __syncthreads();
  }
  if (threadIdx.x == 0 && red[0] != 0.0f) atomicAdd(acc, red[0]);
}

__global__ void finalize(const float* __restrict__ acc, float* __restrict__ out) {
  out[0] = acc[0] / acc[1];
}

// ---------------------------------------------------------------------------
// Orchestration. Workspace: zt 24MB + P12t 12MB + nc14 + dir 2MB + seq + acc
// (~40MB).  neg_cent (64MB) is staged in d_out's attn region and overwritten.
// ---------------------------------------------------------------------------
extern "C" void kernel_launch(void* const* d_in, const int* in_sizes, int n_in,
                              void* d_out, int out_size, void* d_ws, size_t ws_size,
                              hipStream_t stream) {
  const float* m_p    = (const float*)d_in[0];
  const float* logs_p = (const float*)d_in[1];
  const float* z_p    = (const float*)d_in[2];
  const float* logs_q = (const float*)d_in[3];
  const int* textlen  = (const int*)d_in[4];
  const int* mellen   = (const int*)d_in[5];

  float* out  = (float*)d_out;
  float* attn = out + 1;  // [B,T,S]; doubles as neg_cent scratch

  char* ws = (char*)d_ws;
  float* zt   = (float*)ws;  ws += (size_t)Bn * Cn * Tn * 4;
  float* P12t = (float*)ws;  ws += (size_t)Bn * Cn * Sn * 2 * 4;
  float* nc14 = (float*)ws;  ws += (size_t)Bn * Sn * 4;
  unsigned* dir = (unsigned*)ws;  ws += (size_t)Bn * Tn * (Sn / 32) * 4;
  int* seq    = (int*)ws;    ws += (size_t)Bn * Tn * 4;
  float* acc  = (float*)ws;  // acc[0] = kl sum, acc[1] = denom

  hipMemsetAsync(acc, 0, 8, stream);
  prep_stats<<<Bn * Sn, 256, 0, stream>>>(m_p, logs_p, P12t, nc14);
  prep_zt<<<(Bn * Tn * Cn) / 256, 256, 0, stream>>>(z_p, zt);
  negcent_wmma<<<dim3(Sn / 64, Tn / 64, Bn), 128, 0, stream>>>(zt, P12t, nc14, attn);
  mas_forward<<<Bn, Sn, 0, stream>>>(attn, textlen, mellen, dir);
  mas_backward<<<1, 32, 0, stream>>>(textlen, mellen, dir, seq, acc + 1);
  kl_attn<<<dim3(Tn, Bn), 256, 0, stream>>>(m_p, logs_p, z_p, logs_q, seq, attn, acc);
  finalize<<<1, 1, 0, stream>>>(acc, out);
}